// SoftargNet_72962904425014
// MI455X (gfx1250) — compile-verified
//
#include <hip/hip_runtime.h>
#include <hip/hip_bf16.h>

typedef __attribute__((ext_vector_type(2))) float v2f;
typedef __attribute__((ext_vector_type(8))) float v8f;

// Problem constants (fixed by the reference)
static constexpr int B_   = 2;
static constexpr int C_   = 320;
static constexpr int DIM  = 64;
static constexpr int V_   = DIM * DIM * DIM;      // 262144 voxels per batch
static constexpr int GRP_PER_BLK = 8;             // 8 waves/block, 16 voxels/wave
static constexpr int BLK_PER_BATCH_K1 = (V_ / 16) / GRP_PER_BLK;  // 2048
static constexpr int K3_BLK_PER_BATCH = 256;      // each handles 1024 voxels

// Workspace layout (floats)
//  [0, 524288)            corr[b][v]
//  [524288, 528384)       per-block max partials (4096)
//  [528384, 528386)       per-batch max (2)
//  [528386, 530434)       K3 partial sums: 512 blocks * 4 comps

// ---------------------------------------------------------------------------
// K1: pointwise conv (GEMV) via V_WMMA_F32_16X16X4_F32, + per-block max
// ---------------------------------------------------------------------------
__global__ __launch_bounds__(256) void k1_conv_wmma(
    const float* __restrict__ feat, const float* __restrict__ w,
    const float* __restrict__ bias, float* __restrict__ corr,
    float* __restrict__ pmax) {
  __shared__ float sw[C_];
  __shared__ float smax[GRP_PER_BLK];

  const int tid = threadIdx.x;
  for (int i = tid; i < C_; i += 256) sw[i] = w[i];
  __syncthreads();

  const int wave = tid >> 5;
  const int lane = tid & 31;
  const int b    = blockIdx.x / BLK_PER_BATCH_K1;
  const int grp  = (blockIdx.x % BLK_PER_BATCH_K1) * GRP_PER_BLK + wave;
  const size_t v0 = (size_t)grp * 16;
  const float* fb = feat + (size_t)b * C_ * V_;

  // A fragment addressing: lanes 0-15 hold K=kbase..kbase+1 for voxel M=lane&15
  const int m     = lane & 15;
  const int kbase = (lane >> 4) << 1;   // 0 for lanes 0-15, 2 for lanes 16-31

  v8f c = {};
  #pragma unroll 4
  for (int c0 = 0; c0 < C_; c0 += 4) {
    v2f a;
    a.x = fb[(size_t)(c0 + kbase)     * V_ + v0 + m];
    a.y = fb[(size_t)(c0 + kbase + 1) * V_ + v0 + m];
    v2f bw;                              // B[k][n] = w[c0+k], identical over n
    bw.x = sw[c0 + kbase];
    bw.y = sw[c0 + kbase + 1];
    c = __builtin_amdgcn_wmma_f32_16x16x4_f32(
        false, a, false, bw, (short)0, c, false, false);
  }

  const float bv = bias[0];

  // local max over this lane's 8 accumulator rows (all N columns identical)
  float mx = -3.402823466e38f;
  #pragma unroll
  for (int j = 0; j < 8; ++j) mx = fmaxf(mx, c[j] + bv);

  // D column 0: lane 0 holds M=0..7 (voxels v0..v0+7), lane 16 holds M=8..15
  if (lane == 0 || lane == 16) {
    float* dst = corr + (size_t)b * V_ + v0 + (lane == 16 ? 8 : 0);
    #pragma unroll
    for (int j = 0; j < 8; ++j) dst[j] = c[j] + bv;
  }

  // wave-level max reduce (wave32)
  #pragma unroll
  for (int off = 16; off > 0; off >>= 1)
    mx = fmaxf(mx, __shfl_xor(mx, off, 32));
  if (lane == 0) smax[wave] = mx;
  __syncthreads();
  if (tid == 0) {
    float m2 = smax[0];
    #pragma unroll
    for (int i = 1; i < GRP_PER_BLK; ++i) m2 = fmaxf(m2, smax[i]);
    pmax[blockIdx.x] = m2;
  }
}

// ---------------------------------------------------------------------------
// K2: per-batch max over 2048 block partials
// ---------------------------------------------------------------------------
__global__ __launch_bounds__(256) void k2_max(const float* __restrict__ pmax,
                                              float* __restrict__ bmax) {
  __shared__ float s[256];
  const int tid = threadIdx.x;
  for (int b = 0; b < B_; ++b) {
    float m = -3.402823466e38f;
    for (int i = tid; i < BLK_PER_BATCH_K1; i += 256)
      m = fmaxf(m, pmax[b * BLK_PER_BATCH_K1 + i]);
    s[tid] = m;
    __syncthreads();
    for (int st = 128; st > 0; st >>= 1) {
      if (tid < st) s[tid] = fmaxf(s[tid], s[tid + st]);
      __syncthreads();
    }
    if (tid == 0) bmax[b] = s[0];
    __syncthreads();
  }
}

// ---------------------------------------------------------------------------
// K3: exp(corr - max), block partial sums of {e, e*x, e*y, e*z}
// ---------------------------------------------------------------------------
__global__ __launch_bounds__(256) void k3_exp(const float* __restrict__ corr,
                                              const float* __restrict__ bmax,
                                              float* __restrict__ psums) {
  __shared__ float s0[256], s1[256], s2[256], s3[256];
  const int tid = threadIdx.x;
  const int b   = blockIdx.x / K3_BLK_PER_BATCH;
  const int blk = blockIdx.x % K3_BLK_PER_BATCH;
  const int v0  = blk * 1024;
  const float mx = bmax[b];

  float se = 0.f, sx = 0.f, sy = 0.f, sz = 0.f;
  #pragma unroll
  for (int i = 0; i < 4; ++i) {
    const int v = v0 + tid + i * 256;
    const float e = expf(corr[(size_t)b * V_ + v] - mx);
    se += e;
    sx += e * (float)(v & 63);
    sy += e * (float)((v >> 6) & 63);
    sz += e * (float)(v >> 12);
  }
  s0[tid] = se; s1[tid] = sx; s2[tid] = sy; s3[tid] = sz;
  __syncthreads();
  for (int st = 128; st > 0; st >>= 1) {
    if (tid < st) {
      s0[tid] += s0[tid + st];
      s1[tid] += s1[tid + st];
      s2[tid] += s2[tid + st];
      s3[tid] += s3[tid + st];
    }
    __syncthreads();
  }
  if (tid == 0) {
    float* p = psums + (size_t)blockIdx.x * 4;
    p[0] = s0[0]; p[1] = s1[0]; p[2] = s2[0]; p[3] = s3[0];
  }
}

// ---------------------------------------------------------------------------
// K4: final reduction + Mem2Ref transform + write 39 outputs
// ---------------------------------------------------------------------------
__global__ __launch_bounds__(256) void k4_final(
    const float* __restrict__ psums, const float* __restrict__ bounds,
    const int* __restrict__ Zc, const int* __restrict__ Yc,
    const int* __restrict__ Xc, float* __restrict__ out) {
  __shared__ float s[256];
  __shared__ float res[B_][4];
  const int tid = threadIdx.x;

  for (int b = 0; b < B_; ++b) {
    for (int comp = 0; comp < 4; ++comp) {
      s[tid] = psums[(size_t)(b * K3_BLK_PER_BATCH + tid) * 4 + comp];
      __syncthreads();
      for (int st = 128; st > 0; st >>= 1) {
        if (tid < st) s[tid] += s[tid + st];
        __syncthreads();
      }
      if (tid == 0) res[b][comp] = s[0];
      __syncthreads();
    }
  }

  if (tid == 0) {
    out[0] = 0.0f;  // total_loss = 0.0 * smooth_loss
    const float XMIN = bounds[0], XMAX = bounds[1];
    const float YMIN = bounds[2], YMAX = bounds[3];
    const float ZMIN = bounds[4], ZMAX = bounds[5];
    const float zc = (float)Zc[0], yc = (float)Yc[0], xc = (float)Xc[0];
    const float Xm = (float)DIM + 2.f * xc;
    const float Ym = (float)DIM + 2.f * yc;
    const float Zm = (float)DIM + 2.f * zc;
    for (int b = 0; b < B_; ++b) {
      const float se = res[b][0];
      const float x = res[b][1] / se + xc;
      const float y = res[b][2] / se + yc;
      const float z = res[b][3] / se + zc;
      const float xcam = x * ((XMAX - XMIN) / Xm) + XMIN;
      const float ycam = y * ((YMAX - YMIN) / Ym) + YMIN;
      const float zcam = z * ((ZMAX - ZMIN) / Zm) + ZMIN;
      float* o = out + 1 + b * 19;
      o[0] = 2.5f; o[1] = 1.5f; o[2] = 5.0f;   // lens
      for (int i = 0; i < 16; ++i) o[3 + i] = 0.f;
      o[3 + 0]  = 1.f; o[3 + 5]  = 1.f; o[3 + 10] = 1.f; o[3 + 15] = 1.f;
      o[3 + 3]  = xcam; o[3 + 7] = ycam; o[3 + 11] = zcam;
    }
  }
}

// ---------------------------------------------------------------------------
extern "C" void kernel_launch(void* const* d_in, const int* in_sizes, int n_in,
                              void* d_out, int out_size, void* d_ws, size_t ws_size,
                              hipStream_t stream) {
  (void)in_sizes; (void)n_in; (void)out_size; (void)ws_size;
  const float* feat   = (const float*)d_in[0];
  const float* w      = (const float*)d_in[1];
  const float* bias   = (const float*)d_in[2];
  const float* bounds = (const float*)d_in[3];
  const int*   Zc     = (const int*)d_in[4];
  const int*   Yc     = (const int*)d_in[5];
  const int*   Xc     = (const int*)d_in[6];
  float* out = (float*)d_out;

  float* wsf   = (float*)d_ws;
  float* corr  = wsf;                              // 2*262144
  float* pmax  = wsf + (size_t)B_ * V_;            // 4096
  float* bmax  = pmax + B_ * BLK_PER_BATCH_K1;     // 2
  float* psums = bmax + B_;                        // 512*4

  k1_conv_wmma<<<B_ * BLK_PER_BATCH_K1, 256, 0, stream>>>(feat, w, bias, corr, pmax);
  k2_max<<<1, 256, 0, stream>>>(pmax, bmax);
  k3_exp<<<B_ * K3_BLK_PER_BATCH, 256, 0, stream>>>(corr, bmax, psums);
  k4_final<<<1, 256, 0, stream>>>(psums, bounds, Zc, Yc, Xc, out);
}